// KabschLoss_52063593562811
// MI455X (gfx1250) — compile-verified
//
#include <hip/hip_runtime.h>
#include <math.h>

typedef __attribute__((ext_vector_type(2))) float v2f;
typedef __attribute__((ext_vector_type(8))) float v8f;

#define BATCHES 65536
#define NPTS 128
#define WAVES_PER_BLOCK 8
#define WAVE_STRIDE 832   /* floats: xs[0..383] ys[384..767] spill[768..816] pad */

// Per wave: one batch. Build G = [x | y | 1] (128x7) and compute Gram = G^T G
// with v_wmma_f32_16x16x4_f32 accumulating 4 points per step.
// Gram contains every statistic needed:
//   Gram[j][k]   (j,k<3)  = sum x_j x_k      -> sum|x|^2 on diagonal
//   Gram[3+j][3+k]        = sum y_j y_k      -> sum|y|^2 on diagonal
//   Gram[j][3+k]          = sum x_j y_k      -> raw cross covariance
//   Gram[j][6], Gram[3+k][6]                 -> sum x_j, sum y_k (means)
// Batch loss = sum|xc|^2 + sum|yc|^2 - 2 * nuclear_norm(C), C = centered covariance.
__global__ __launch_bounds__(256)
void kabsch_main(const float* __restrict__ x, const float* __restrict__ y,
                 double* __restrict__ accum) {
  __shared__ float lds[WAVES_PER_BLOCK * WAVE_STRIDE];
  const int tid  = threadIdx.x;
  const int lane = tid & 31;
  const int wv   = tid >> 5;
  const int batch = blockIdx.x * WAVES_PER_BLOCK + wv;

  float* wbase = &lds[wv * WAVE_STRIDE];

  // ---- stage this batch's x,y into LDS: 3 float4 loads each per lane ----
  const float4* xb = (const float4*)(x + (size_t)batch * (NPTS * 3));
  const float4* yb = (const float4*)(y + (size_t)batch * (NPTS * 3));
  float4* xs4 = (float4*)wbase;
  float4* ys4 = (float4*)(wbase + 384);
#pragma unroll
  for (int i = 0; i < 3; ++i) {
    xs4[lane * 3 + i] = xb[lane * 3 + i];
    ys4[lane * 3 + i] = yb[lane * 3 + i];
  }
  __syncthreads();

  // ---- per-lane operand addressing ----
  // A-matrix 16x4 f32 layout: M = lane%16, VGPR0 holds K=(0|2), VGPR1 holds K=(1|3)
  // by half-wave. A[m][k] = G[point k][coord m]; B[k][n] = G[point k][coord n] has
  // the transposed layout, so the same register pair feeds both operands.
  const int  c    = lane & 15;        // coordinate index (0..2 x, 3..5 y, 6 ones)
  const int  half = lane >> 4;
  const int  coff = (c < 3) ? c : ((c < 6) ? (384 + (c - 3)) : 0);
  const float cval = (c == 6) ? 1.0f : 0.0f;
  const bool useMem = (c < 6);
  const float* gp = wbase + coff + half * 6;   // half selects points +2 (6 floats)

  v8f acc = {};
#pragma unroll
  for (int kk = 0; kk < 32; ++kk) {            // 4 points per WMMA, 128 total
    float v0 = gp[kk * 12 + 0];                // point kk*4 + 2*half
    float v1 = gp[kk * 12 + 3];                // point kk*4 + 2*half + 1
    v2f a;
    a.x = useMem ? v0 : cval;
    a.y = useMem ? v1 : cval;
    // D = A x B + C  (A == B register data; Gram accumulate)
    acc = __builtin_amdgcn_wmma_f32_16x16x4_f32(false, a, false, a,
                                                (short)0, acc, false, false);
  }

  // ---- spill 7x7 Gram corner: lanes 0..6 hold columns, acc[j] = row j ----
  float* spill = wbase + 768;
  if (lane < 7) {
#pragma unroll
    for (int j = 0; j < 7; ++j) spill[lane * 7 + j] = acc[j];
  }
  __syncthreads();

  if (lane == 0) {
    float g[7][7];
#pragma unroll
    for (int n = 0; n < 7; ++n)
#pragma unroll
      for (int j = 0; j < 7; ++j) g[j][n] = spill[n * 7 + j];

    const float invN = 1.0f / (float)NPTS;
    const float sx0 = g[0][6], sx1 = g[1][6], sx2 = g[2][6];
    const float sy0 = g[3][6], sy1 = g[4][6], sy2 = g[5][6];
    const float mx0 = sx0 * invN, mx1 = sx1 * invN, mx2 = sx2 * invN;
    const float my0 = sy0 * invN, my1 = sy1 * invN, my2 = sy2 * invN;

    // centered covariance C[j][k] = sum x_j y_k - N mx_j my_k
    const float C00 = g[0][3] - sx0 * my0, C01 = g[0][4] - sx0 * my1, C02 = g[0][5] - sx0 * my2;
    const float C10 = g[1][3] - sx1 * my0, C11 = g[1][4] - sx1 * my1, C12 = g[1][5] - sx1 * my2;
    const float C20 = g[2][3] - sx2 * my0, C21 = g[2][4] - sx2 * my1, C22 = g[2][5] - sx2 * my2;

    const float sxx = (g[0][0] + g[1][1] + g[2][2]) - (sx0 * mx0 + sx1 * mx1 + sx2 * mx2);
    const float syy = (g[3][3] + g[4][4] + g[5][5]) - (sy0 * my0 + sy1 * my1 + sy2 * my2);

    // Bm = C^T C (symmetric PSD); nuclear norm = sum sqrt(eig(Bm))
    const float b00 = C00 * C00 + C10 * C10 + C20 * C20;
    const float b11 = C01 * C01 + C11 * C11 + C21 * C21;
    const float b22 = C02 * C02 + C12 * C12 + C22 * C22;
    const float b01 = C00 * C01 + C10 * C11 + C20 * C21;
    const float b02 = C00 * C02 + C10 * C12 + C20 * C22;
    const float b12 = C01 * C02 + C11 * C12 + C21 * C22;

    const float p1 = b01 * b01 + b02 * b02 + b12 * b12;
    const float q  = (b00 + b11 + b22) * (1.0f / 3.0f);
    const float d0 = b00 - q, d1 = b11 - q, d2 = b22 - q;
    const float p2 = d0 * d0 + d1 * d1 + d2 * d2 + 2.0f * p1;
    const float p  = sqrtf(fmaxf(p2, 0.0f) * (1.0f / 6.0f));

    float e0, e1, e2;
    if (p < 1e-18f) {
      e0 = e1 = e2 = q;
    } else {
      const float inv = 1.0f / p;
      const float m00 = d0 * inv, m11 = d1 * inv, m22 = d2 * inv;
      const float m01 = b01 * inv, m02 = b02 * inv, m12 = b12 * inv;
      float det = m00 * (m11 * m22 - m12 * m12)
                - m01 * (m01 * m22 - m12 * m02)
                + m02 * (m01 * m12 - m11 * m02);
      float r = fminf(1.0f, fmaxf(-1.0f, det * 0.5f));
      const float phi = acosf(r) * (1.0f / 3.0f);
      e0 = q + 2.0f * p * cosf(phi);
      e2 = q + 2.0f * p * cosf(phi + 2.0943951023931953f);  // + 2*pi/3
      e1 = 3.0f * q - e0 - e2;
    }
    const float nuc = sqrtf(fmaxf(e0, 0.0f)) + sqrtf(fmaxf(e1, 0.0f)) + sqrtf(fmaxf(e2, 0.0f));
    const float loss = sxx + syy - 2.0f * nuc;
    atomicAdd(accum, (double)loss);
  }
}

__global__ void kabsch_init(double* a) { *a = 0.0; }

__global__ void kabsch_fin(const double* __restrict__ a, float* __restrict__ out) {
  out[0] = (float)(*a / ((double)BATCHES * (double)NPTS * 3.0));
}

extern "C" void kernel_launch(void* const* d_in, const int* in_sizes, int n_in,
                              void* d_out, int out_size, void* d_ws, size_t ws_size,
                              hipStream_t stream) {
  const float* x = (const float*)d_in[0];
  const float* y = (const float*)d_in[1];
  float* out = (float*)d_out;
  double* accum = (double*)d_ws;

  kabsch_init<<<1, 1, 0, stream>>>(accum);
  kabsch_main<<<BATCHES / WAVES_PER_BLOCK, 256, 0, stream>>>(x, y, accum);
  kabsch_fin<<<1, 1, 0, stream>>>(accum, out);
}